// TitansMACMemory_5961414607218
// MI455X (gfx1250) — compile-verified
//
#include <hip/hip_runtime.h>
#include <hip/hip_bf16.h>

// ---------------------------------------------------------------------------
// Titans MAC memory — MI455X (gfx1250, wave32) implementation.
//   T=512, B=32, H=1024, M=64
// GEMMs run on V_WMMA_F32_16X16X4_F32 (fp32 tensor op — keeps reference fp32
// precision). kv GEMM stages its A strip through LDS with
// GLOBAL_LOAD_ASYNC_TO_LDS_B128 (ASYNCcnt-tracked, double-buffered) since all
// 4 waves of a block share the same 16 A rows. Sequential 512-step recurrence
// runs as one persistent workgroup (latency-bound; state is L2-resident),
// with 4-way split accumulators to break FMA dependency chains.
// ---------------------------------------------------------------------------

typedef __attribute__((ext_vector_type(2))) float v2f;
typedef __attribute__((ext_vector_type(8))) float v8f;

#define T_DIM 512
#define B_DIM 32
#define H_DIM 1024
#define M_DIM 64
#define ROWS  (T_DIM * B_DIM)      // 16384 token rows

__device__ __forceinline__ v8f wmma_f32_k4(v2f a, v2f b, v8f c) {
  // D(16x16,f32) = A(16x4,f32) x B(4x16,f32) + C
  return __builtin_amdgcn_wmma_f32_16x16x4_f32(
      /*neg_a=*/false, a, /*neg_b=*/false, b,
      /*c_mod=*/(short)0, c, /*reuse_a=*/false, /*reuse_b=*/false);
}

__device__ __forceinline__ float gelu_exact(float x) {
  return 0.5f * x * (1.0f + erff(x * 0.70710678118654752f));
}

// ---------------------------------------------------------------------------
// Kernel 1: k = x @ W_K^T, v = x @ W_V^T   (C[i,j] = sum_h A[i,h]*W[j,h])
// Block = 128 thr (4 waves) -> one 16-row A strip, 4 j-tiles. A strip is
// staged cooperatively into LDS via async-to-LDS B128 (double buffered,
// 64-wide K chunks); each wave then runs 2 WMMAs (k and v) per K-step with
// the A fragment read from LDS and W rows loaded directly.
// ---------------------------------------------------------------------------
__global__ __launch_bounds__(128) void kv_gemm_kernel(
    const float* __restrict__ x, const float* __restrict__ Wk,
    const float* __restrict__ Wv, float* __restrict__ kout,
    float* __restrict__ vout) {
  __shared__ __align__(16) float a_s[2][16][64];   // 2 x 4KB K-chunks

  const int tid  = threadIdx.x;
  const int wave = tid >> 5;
  const int lane = tid & 31;
  const int l    = lane & 15;
  const int half = lane >> 4;

  const int i0 = (blockIdx.x >> 4) * 16;
  const int j0 = ((blockIdx.x & 15) * 4 + wave) * 16;

  const float* abase = x  + (size_t)i0 * H_DIM;
  const float* bkr   = Wk + (size_t)(j0 + l) * H_DIM + 2 * half;
  const float* bvr   = Wv + (size_t)(j0 + l) * H_DIM + 2 * half;

  // Cooperative staging: 16x64 f32 chunk = 4KB = two b128 per thread.
  const int srow = tid >> 4;            // 0..7
  const int scol = (tid & 15) * 4;      // 0,4,...,60
  const float* g0 = abase + (size_t)srow * H_DIM + scol;
  const float* g1 = abase + (size_t)(srow + 8) * H_DIM + scol;
  // Low 32 bits of a flat LDS pointer are the wave-relative LDS byte offset
  // (ISA 10.2: LDS aperture truncates to addr[31:0]).
  const unsigned lbase = (unsigned)(size_t)(&a_s[0][0][0]);
  const unsigned loff0 = (unsigned)(tid * 16);
  const unsigned loff1 = (unsigned)(2048 + tid * 16);

  {  // prologue: stage chunk 0 into buffer 0
    unsigned d0 = lbase + loff0, d1 = lbase + loff1;
    asm volatile("global_load_async_to_lds_b128 %0, %1, off"
                 :: "v"(d0), "v"(g0) : "memory");
    asm volatile("global_load_async_to_lds_b128 %0, %1, off"
                 :: "v"(d1), "v"(g1) : "memory");
  }

  v8f acck = {};
  v8f accv = {};
  for (int c = 0; c < H_DIM / 64; ++c) {
    if (c + 1 < H_DIM / 64) {
      // stream next chunk into the other buffer, then retire chunk c
      const unsigned boff = ((c + 1) & 1) ? 4096u : 0u;
      unsigned d0 = lbase + boff + loff0, d1 = lbase + boff + loff1;
      const float* n0 = g0 + (size_t)(c + 1) * 64;
      const float* n1 = g1 + (size_t)(c + 1) * 64;
      asm volatile("global_load_async_to_lds_b128 %0, %1, off"
                   :: "v"(d0), "v"(n0) : "memory");
      asm volatile("global_load_async_to_lds_b128 %0, %1, off"
                   :: "v"(d1), "v"(n1) : "memory");
      // async loads complete in order: <=2 outstanding => chunk c landed
      asm volatile("s_wait_asynccnt 2" ::: "memory");
    } else {
      asm volatile("s_wait_asynccnt 0" ::: "memory");
    }
    __syncthreads();

    const int kb = c * 64;
    const float* asrc = &a_s[c & 1][0][0];
    __builtin_prefetch(bkr + kb + 256, 0, 3);   // global_prefetch_b8
    __builtin_prefetch(bvr + kb + 256, 0, 3);
#pragma unroll
    for (int kk = 0; kk < 64; kk += 4) {
      v2f a  = *(const v2f*)(asrc + l * 64 + kk + 2 * half);   // ds_load_b64
      v2f bk = *(const v2f*)(bkr + kb + kk);
      v2f bv = *(const v2f*)(bvr + kb + kk);
      acck = wmma_f32_k4(a, bk, acck);
      accv = wmma_f32_k4(a, bv, accv);
    }
    __syncthreads();   // everyone done reading buffer (c&1) before reuse
  }

#pragma unroll
  for (int r = 0; r < 8; ++r) {
    size_t off = (size_t)(i0 + r + 8 * half) * H_DIM + (j0 + l);
    kout[off] = acck[r];
    vout[off] = accv[r];
  }
}

// ---------------------------------------------------------------------------
// Kernel 2: pred = gelu(k @ W1^T) @ W2^T ; reads = pred ;
//           assoc += sum((pred - v)^2) / (B*H)
// ---------------------------------------------------------------------------
__global__ __launch_bounds__(128) void pred_kernel(
    const float* __restrict__ kin, const float* __restrict__ vin,
    const float* __restrict__ W1, const float* __restrict__ W2,
    float* __restrict__ reads, float* __restrict__ assoc) {
  __shared__ float h_s[16][M_DIM + 4];   // +4 pad (even -> keeps v2f align)
  __shared__ float red[128];

  const int wave = threadIdx.x >> 5;
  const int lane = threadIdx.x & 31;
  const int l    = lane & 15;
  const int half = lane >> 4;
  const int i0   = blockIdx.x * 16;

  // ---- phase 1: h strip (16 x 64), wave w owns cols [16w, 16w+16) ----
  {
    const int jh0 = wave * 16;
    const float* arow = kin + (size_t)(i0 + l) * H_DIM + 2 * half;
    const float* brow = W1  + (size_t)(jh0 + l) * H_DIM + 2 * half;
    v8f acc = {};
    for (int kb = 0; kb < H_DIM; kb += 64) {
      __builtin_prefetch(arow + kb + 256, 0, 3);
#pragma unroll
      for (int kk = 0; kk < 64; kk += 4) {
        const int k0 = kb + kk;
        v2f a = *(const v2f*)(arow + k0);
        v2f b = *(const v2f*)(brow + k0);
        acc = wmma_f32_k4(a, b, acc);
      }
    }
#pragma unroll
    for (int r = 0; r < 8; ++r)
      h_s[r + 8 * half][jh0 + l] = gelu_exact(acc[r]);
  }
  __syncthreads();

  // ---- phase 2: pred tiles, K = 64 ----
  float sumsq = 0.0f;
  for (int jt = wave; jt < 64; jt += 4) {
    const int j0 = jt * 16;
    const float* brow = W2 + (size_t)(j0 + l) * M_DIM + 2 * half;
    v8f acc = {};
#pragma unroll
    for (int k0 = 0; k0 < M_DIM; k0 += 4) {
      v2f a = *(const v2f*)(&h_s[l][k0 + 2 * half]);
      v2f b = *(const v2f*)(brow + k0);
      acc = wmma_f32_k4(a, b, acc);
    }
#pragma unroll
    for (int r = 0; r < 8; ++r) {
      size_t off = (size_t)(i0 + r + 8 * half) * H_DIM + (j0 + l);
      float p = acc[r];
      reads[off] = p;
      float d = p - vin[off];
      sumsq += d * d;
    }
  }

  red[threadIdx.x] = sumsq;
  __syncthreads();
  for (int s = 64; s > 0; s >>= 1) {
    if (threadIdx.x < s) red[threadIdx.x] += red[threadIdx.x + s];
    __syncthreads();
  }
  if (threadIdx.x == 0)
    atomicAdd(assoc, red[0] * (1.0f / (float)(B_DIM * H_DIM)));
}

// ---------------------------------------------------------------------------
// Kernel 3: sequential TTT recurrence, one persistent workgroup (32 waves).
// Latency-bound: all long reductions use 4 split accumulators for ILP.
// ---------------------------------------------------------------------------
__global__ __launch_bounds__(1024) void scan_kernel(
    const float* __restrict__ x, const float* __restrict__ kbuf,
    const float* __restrict__ vbuf,
    const float* __restrict__ w_eta, const float* __restrict__ b_eta,
    const float* __restrict__ w_theta, const float* __restrict__ b_theta,
    const float* __restrict__ w_alpha, const float* __restrict__ b_alpha,
    float* __restrict__ W1c, float* __restrict__ W2c,
    float* __restrict__ m1c, float* __restrict__ m2c,
    float* __restrict__ g1, float* __restrict__ g2, float* __restrict__ dp,
    float* __restrict__ surprises, float* __restrict__ W1out,
    float* __restrict__ W2out) {
  const int tid  = threadIdx.x;
  const int lane = tid & 31;
  const int wave = tid >> 5;          // wave == batch index b (32 waves)

  __shared__ float hp[B_DIM][M_DIM + 1];
  __shared__ float gh[B_DIM][M_DIM + 1];
  __shared__ float dhp[B_DIM][M_DIM + 1];
  __shared__ float gE[B_DIM], gT[B_DIM], gA[B_DIM];
  __shared__ float red[1024];
  __shared__ float sc[8];   // 0=eta 1=theta 2=alpha 3=ss1 4=ss2 5=s1 6=s2

  for (int t = 0; t < T_DIM; ++t) {
    const float* tokT = x    + (size_t)t * B_DIM * H_DIM;
    const float* ktT  = kbuf + (size_t)t * B_DIM * H_DIM;
    const float* vtT  = vbuf + (size_t)t * B_DIM * H_DIM;

    // ---- gates: eta/theta/alpha = mean_b sigmoid(tok[b]·w + b) ----
    {
      const float* tok = tokT + (size_t)wave * H_DIM;
      float se = 0.f, st = 0.f, sa = 0.f;
      for (int h = lane; h < H_DIM; h += 32) {
        float xv = tok[h];
        se += xv * w_eta[h];
        st += xv * w_theta[h];
        sa += xv * w_alpha[h];
      }
      for (int o = 16; o > 0; o >>= 1) {
        se += __shfl_xor(se, o);
        st += __shfl_xor(st, o);
        sa += __shfl_xor(sa, o);
      }
      if (lane == 0) {
        gE[wave] = 1.0f / (1.0f + expf(-(se + b_eta[0])));
        gT[wave] = 1.0f / (1.0f + expf(-(st + b_theta[0])));
        gA[wave] = 1.0f / (1.0f + expf(-(sa + b_alpha[0])));
      }
    }
    __syncthreads();
    if (tid == 0) {
      float e = 0.f, th = 0.f, al = 0.f;
      for (int b = 0; b < B_DIM; ++b) { e += gE[b]; th += gT[b]; al += gA[b]; }
      sc[0] = e * (1.0f / B_DIM);
      sc[1] = th * (1.0f / B_DIM);
      sc[2] = al * (1.0f / B_DIM);
    }
    __syncthreads();

    // ---- forward: hp[b][m] = kt[b]·W1c[m], gh = gelu(hp) ----
    for (int idx = tid; idx < B_DIM * M_DIM; idx += 1024) {
      int b = idx >> 6, m = idx & 63;
      const float* kr = ktT + (size_t)b * H_DIM;
      const float* wr = W1c + (size_t)m * H_DIM;
      float s0 = 0.f, s1 = 0.f, s2 = 0.f, s3 = 0.f;
      for (int h = 0; h < H_DIM; h += 4) {
        s0 = fmaf(kr[h + 0], wr[h + 0], s0);
        s1 = fmaf(kr[h + 1], wr[h + 1], s1);
        s2 = fmaf(kr[h + 2], wr[h + 2], s2);
        s3 = fmaf(kr[h + 3], wr[h + 3], s3);
      }
      float s = (s0 + s1) + (s2 + s3);
      hp[b][m] = s;
      gh[b][m] = gelu_exact(s);
    }
    __syncthreads();

    // ---- dp[b,j] = (gh[b]·W2c[j] - vt[b,j]) * 2/(B*H) ----
    for (int idx = tid; idx < B_DIM * H_DIM; idx += 1024) {
      int b = idx >> 10, j = idx & 1023;
      const float* w2r = W2c + (size_t)j * M_DIM;
      float s0 = 0.f, s1 = 0.f, s2 = 0.f, s3 = 0.f;
      for (int m = 0; m < M_DIM; m += 4) {
        s0 = fmaf(gh[b][m + 0], w2r[m + 0], s0);
        s1 = fmaf(gh[b][m + 1], w2r[m + 1], s1);
        s2 = fmaf(gh[b][m + 2], w2r[m + 2], s2);
        s3 = fmaf(gh[b][m + 3], w2r[m + 3], s3);
      }
      dp[idx] = ((s0 + s1) + (s2 + s3) - vtT[idx]) *
                (2.0f / (float)(B_DIM * H_DIM));
    }
    __syncthreads();

    // ---- g2[j,m] = sum_b dp[b,j] * gh[b,m]  (+ ||g2||^2) ----
    float ss2 = 0.f;
    for (int idx = tid; idx < H_DIM * M_DIM; idx += 1024) {
      int j = idx >> 6, m = idx & 63;
      float s0 = 0.f, s1 = 0.f, s2 = 0.f, s3 = 0.f;
      for (int b = 0; b < B_DIM; b += 4) {
        s0 = fmaf(dp[(b + 0) * H_DIM + j], gh[b + 0][m], s0);
        s1 = fmaf(dp[(b + 1) * H_DIM + j], gh[b + 1][m], s1);
        s2 = fmaf(dp[(b + 2) * H_DIM + j], gh[b + 2][m], s2);
        s3 = fmaf(dp[(b + 3) * H_DIM + j], gh[b + 3][m], s3);
      }
      float s = (s0 + s1) + (s2 + s3);
      g2[idx] = s;
      ss2 = fmaf(s, s, ss2);
    }

    // ---- dhp[b,m] = (dp[b]·W2c[:,m]) * gelu'(hp[b,m]) ----
    for (int idx = tid; idx < B_DIM * M_DIM; idx += 1024) {
      int b = idx >> 6, m = idx & 63;
      const float* dpr = dp + (size_t)b * H_DIM;
      float s0 = 0.f, s1 = 0.f, s2 = 0.f, s3 = 0.f;
      for (int j = 0; j < H_DIM; j += 4) {
        s0 = fmaf(dpr[j + 0], W2c[(j + 0) * M_DIM + m], s0);
        s1 = fmaf(dpr[j + 1], W2c[(j + 1) * M_DIM + m], s1);
        s2 = fmaf(dpr[j + 2], W2c[(j + 2) * M_DIM + m], s2);
        s3 = fmaf(dpr[j + 3], W2c[(j + 3) * M_DIM + m], s3);
      }
      float s   = (s0 + s1) + (s2 + s3);
      float hx  = hp[b][m];
      float cdf = 0.5f * (1.0f + erff(hx * 0.70710678118654752f));
      float pdf = expf(-0.5f * hx * hx) * 0.39894228040143268f;
      dhp[b][m] = s * (cdf + hx * pdf);
    }
    __syncthreads();

    // ---- g1[m,h] = sum_b dhp[b,m] * kt[b,h]  (+ ||g1||^2) ----
    float ss1 = 0.f;
    for (int idx = tid; idx < M_DIM * H_DIM; idx += 1024) {
      int m = idx >> 10, h = idx & 1023;
      const float* kr = ktT + h;
      float s0 = 0.f, s1 = 0.f, s2 = 0.f, s3 = 0.f;
      for (int b = 0; b < B_DIM; b += 4) {
        s0 = fmaf(dhp[b + 0][m], kr[(b + 0) * H_DIM], s0);
        s1 = fmaf(dhp[b + 1][m], kr[(b + 1) * H_DIM], s1);
        s2 = fmaf(dhp[b + 2][m], kr[(b + 2) * H_DIM], s2);
        s3 = fmaf(dhp[b + 3][m], kr[(b + 3) * H_DIM], s3);
      }
      float s = (s0 + s1) + (s2 + s3);
      g1[idx] = s;
      ss1 = fmaf(s, s, ss1);
    }

    // ---- block-reduce the two squared norms ----
    red[tid] = ss1;
    __syncthreads();
    for (int s = 512; s > 0; s >>= 1) {
      if (tid < s) red[tid] += red[tid + s];
      __syncthreads();
    }
    if (tid == 0) sc[3] = red[0];
    __syncthreads();
    red[tid] = ss2;
    __syncthreads();
    for (int s = 512; s > 0; s >>= 1) {
      if (tid < s) red[tid] += red[tid + s];
      __syncthreads();
    }
    if (tid == 0) sc[4] = red[0];
    __syncthreads();

    if (tid == 0) {
      float n1 = sqrtf(sc[3]), n2 = sqrtf(sc[4]);
      surprises[t] = 0.5f * (n1 + n2);
      float sf1 = fmaxf(n1, 1e-8f), sf2 = fmaxf(n2, 1e-8f);
      sc[5] = (n1 < 1e-8f) ? 1.0f : tanhf(sf1) / sf1;   // max_norm = 1
      sc[6] = (n2 < 1e-8f) ? 1.0f : tanhf(sf2) / sf2;
    }
    __syncthreads();

    // ---- momentum + weight updates ----
    {
      float eta = sc[0], theta = sc[1], oma = 1.0f - sc[2];
      float s1 = sc[5], s2v = sc[6];
      for (int idx = tid; idx < M_DIM * H_DIM; idx += 1024) {
        float mn1 = eta * m1c[idx] - theta * (g1[idx] * s1);
        m1c[idx] = mn1;
        W1c[idx] = oma * W1c[idx] + mn1;
        float mn2 = eta * m2c[idx] - theta * (g2[idx] * s2v);
        m2c[idx] = mn2;
        W2c[idx] = oma * W2c[idx] + mn2;
      }
    }
    __syncthreads();
  }

  for (int idx = tid; idx < M_DIM * H_DIM; idx += 1024) {
    W1out[idx] = W1c[idx];
    W2out[idx] = W2c[idx];
  }
}

// ---------------------------------------------------------------------------
extern "C" void kernel_launch(void* const* d_in, const int* in_sizes, int n_in,
                              void* d_out, int out_size, void* d_ws,
                              size_t ws_size, hipStream_t stream) {
  (void)in_sizes; (void)n_in; (void)out_size; (void)ws_size;
  const float* x       = (const float*)d_in[0];
  const float* W_K     = (const float*)d_in[1];
  const float* W_V     = (const float*)d_in[2];
  const float* w_eta   = (const float*)d_in[3];
  const float* b_eta   = (const float*)d_in[4];
  const float* w_theta = (const float*)d_in[5];
  const float* b_theta = (const float*)d_in[6];
  const float* w_alpha = (const float*)d_in[7];
  const float* b_alpha = (const float*)d_in[8];
  const float* W1      = (const float*)d_in[9];
  const float* W2      = (const float*)d_in[10];
  const float* m1      = (const float*)d_in[11];
  const float* m2      = (const float*)d_in[12];

  const size_t R  = (size_t)T_DIM * B_DIM * H_DIM;   // 16777216
  const size_t MH = (size_t)M_DIM * H_DIM;           // 65536

  float* out    = (float*)d_out;
  float* reads  = out;
  float* assoc  = out + R;
  float* surpr  = assoc + 1;
  float* W1out  = surpr + T_DIM;
  float* W2out  = W1out + MH;

  float* ws   = (float*)d_ws;
  float* kbuf = ws;
  float* vbuf = kbuf + R;
  float* W1c  = vbuf + R;
  float* W2c  = W1c + MH;
  float* m1c  = W2c + MH;
  float* m2c  = m1c + MH;
  float* g1   = m2c + MH;
  float* g2   = g1 + MH;
  float* dpb  = g2 + MH;

  hipMemsetAsync(assoc, 0, sizeof(float), stream);
  hipMemcpyAsync(W1c, W1, MH * sizeof(float), hipMemcpyDeviceToDevice, stream);
  hipMemcpyAsync(W2c, W2, MH * sizeof(float), hipMemcpyDeviceToDevice, stream);
  hipMemcpyAsync(m1c, m1, MH * sizeof(float), hipMemcpyDeviceToDevice, stream);
  hipMemcpyAsync(m2c, m2, MH * sizeof(float), hipMemcpyDeviceToDevice, stream);

  kv_gemm_kernel<<<16384, 128, 0, stream>>>(x, W_K, W_V, kbuf, vbuf);
  pred_kernel<<<1024, 128, 0, stream>>>(kbuf, vbuf, W1, W2, reads, assoc);
  scan_kernel<<<1, 1024, 0, stream>>>(x, kbuf, vbuf, w_eta, b_eta, w_theta,
                                      b_theta, w_alpha, b_alpha, W1c, W2c, m1c,
                                      m2c, g1, g2, dpb, surpr, W1out, W2out);
}